// CompetitiveNetwork_85736137163122
// MI455X (gfx1250) — compile-verified
//
#include <hip/hip_runtime.h>

typedef __attribute__((ext_vector_type(2))) float v2f;
typedef __attribute__((ext_vector_type(8))) float v8f;

#define N_LOOP 21  // reference: 20 fori iters + (BF,AF,BF) epilogue == 21*(S,T) + final S

__device__ __forceinline__ float fast_rcp(float x) {
  return __builtin_amdgcn_rcpf(x);
}

// One 16x16 output tile of the transposed mat-vec: D = A_tile x X^T, where the
// contraction index is permuted by pi so the B-operand fragment for k-step k is
// exactly the resident T-layout register pair X[k>>2][k&3] (no cross-lane moves).
__device__ __forceinline__ v8f matvec_tile(const v2f fA[16], const v2f X[4][4]) {
  v8f c = {0.f, 0.f, 0.f, 0.f, 0.f, 0.f, 0.f, 0.f};
#pragma unroll
  for (int k = 0; k < 16; ++k)
    c = __builtin_amdgcn_wmma_f32_16x16x4_f32(false, fA[k], false, X[k >> 2][k & 3],
                                              (short)0, c, false, false);
  return c;
}

__global__ __launch_bounds__(256) void comp_net_kernel(
    const float* __restrict__ AT, const float* __restrict__ K_raw,
    const float* __restrict__ BT_raw, const float* __restrict__ W_raw,
    const float* __restrict__ b_raw, float* __restrict__ Y) {
  __shared__ float sKT[64 * 64];  // Kexp^T  (A-operand of S-phase)
  __shared__ float sK[64 * 64];   // Kexp    (A-operand of T-phase)
  __shared__ float sM[64 * 64];   // Kexp * clip(W)  (A-operand of readout)

  const int tid = threadIdx.x;
  // Block-cooperative LDS setup: Kexp = clip(exp(K_raw),0,1000); M = Kexp*clip(W,-10,10)
  for (int idx = tid; idx < 4096; idx += 256) {
    const int i = idx >> 6, j = idx & 63;
    const float ke = fminf(expf(K_raw[idx]), 1000.0f);
    sK[idx] = ke;
    sKT[j * 64 + i] = ke;
    float w = W_raw[idx];
    w = fminf(fmaxf(w, -10.0f), 10.0f);
    sM[idx] = ke * w;
  }
  __syncthreads();

  const int lane = tid & 31;
  const int hi = (lane >> 4) & 1;
  const int lr = lane & 15;
  const int warp = tid >> 5;
  const int row = ((int)blockIdx.x * 8 + warp) * 16 + lr;  // batch row for this lane

  // Register-resident A-operand fragments of Kexp^T and Kexp, in pi-permuted
  // K order: fragment (mtA, k) holds cols (16*(k>>2)+2*(k&3)+8*hi, +1) of row 16*mtA+lr.
  v2f fKT[4][16], fK[4][16];
#pragma unroll
  for (int mtA = 0; mtA < 4; ++mtA) {
    const float* rT = sKT + (16 * mtA + lr) * 64 + 8 * hi;
    const float* rK = sK + (16 * mtA + lr) * 64 + 8 * hi;
#pragma unroll
    for (int k = 0; k < 16; ++k) {
      const int col = 16 * (k >> 2) + 2 * (k & 3);
      fKT[mtA][k] = *(const v2f*)(rT + col);
      fK[mtA][k] = *(const v2f*)(rK + col);
    }
  }

  // Per-lane T-layout state as v2f pairs: X[mt][q] = (n = 16mt+2q+8hi, n+1)
  v2f ATt[4][4], AFt[4][4], BFt[4][4], BTv[4][4];
#pragma unroll
  for (int mt = 0; mt < 4; ++mt) {
    const float* src = AT + row * 64 + 16 * mt + 8 * hi;
    const float* bsrc = BT_raw + 16 * mt + 8 * hi;
#pragma unroll
    for (int q = 0; q < 4; ++q) {
      ATt[mt][q] = *(const v2f*)(src + 2 * q);
      AFt[mt][q] = ATt[mt][q];  // AF starts as ATe
      v2f t;
      t[0] = fminf(expf(bsrc[2 * q]), 1000.0f);
      t[1] = fminf(expf(bsrc[2 * q + 1]), 1000.0f);
      BTv[mt][q] = t;
    }
  }

  v8f acc[4];

  for (int it = 0; it < N_LOOP; ++it) {
    // S-phase: s = K^T a  (as S^T = Kexp^T @ AF^T), bf = BT * rcp(s+1)
#pragma unroll
    for (int mtA = 0; mtA < 4; ++mtA) acc[mtA] = matvec_tile(fKT[mtA], AFt);
#pragma unroll
    for (int mt = 0; mt < 4; ++mt)
#pragma unroll
      for (int q = 0; q < 4; ++q) {
        v2f t;
        t[0] = BTv[mt][q][0] * fast_rcp(acc[mt][2 * q] + 1.0f);
        t[1] = BTv[mt][q][1] * fast_rcp(acc[mt][2 * q + 1] + 1.0f);
        BFt[mt][q] = t;
      }

    // T-phase: t = K bf  (as T^T = Kexp @ BF^T), a = AT * rcp(t+1)
#pragma unroll
    for (int mtA = 0; mtA < 4; ++mtA) acc[mtA] = matvec_tile(fK[mtA], BFt);
#pragma unroll
    for (int mt = 0; mt < 4; ++mt)
#pragma unroll
      for (int q = 0; q < 4; ++q) {
        v2f t;
        t[0] = ATt[mt][q][0] * fast_rcp(acc[mt][2 * q] + 1.0f);
        t[1] = ATt[mt][q][1] * fast_rcp(acc[mt][2 * q + 1] + 1.0f);
        AFt[mt][q] = t;
      }
  }

  // Final S-phase -> BF
#pragma unroll
  for (int mtA = 0; mtA < 4; ++mtA) acc[mtA] = matvec_tile(fKT[mtA], AFt);
#pragma unroll
  for (int mt = 0; mt < 4; ++mt)
#pragma unroll
    for (int q = 0; q < 4; ++q) {
      v2f t;
      t[0] = BTv[mt][q][0] * fast_rcp(acc[mt][2 * q] + 1.0f);
      t[1] = BTv[mt][q][1] * fast_rcp(acc[mt][2 * q + 1] + 1.0f);
      BFt[mt][q] = t;
    }

  // Readout: U = M @ bf; Y[b] = dot(a, U) + clip(b0). M fragments used once -> from LDS.
  float part = 0.0f;
#pragma unroll
  for (int mtA = 0; mtA < 4; ++mtA) {
    v2f fM[16];
    const float* rM = sM + (16 * mtA + lr) * 64 + 8 * hi;
#pragma unroll
    for (int k = 0; k < 16; ++k)
      fM[k] = *(const v2f*)(rM + 16 * (k >> 2) + 2 * (k & 3));
    const v8f u = matvec_tile(fM, BFt);
#pragma unroll
    for (int q = 0; q < 4; ++q) {
      part = fmaf(AFt[mtA][q][0], u[2 * q], part);
      part = fmaf(AFt[mtA][q][1], u[2 * q + 1], part);
    }
  }
  part += __shfl_xor(part, 16, 32);  // lanes l and l+16 hold complementary i-halves of row b

  if (hi == 0) {
    const float bc = fminf(fmaxf(b_raw[0], -10.0f), 10.0f);
    Y[row] = part + bc;
  }
}

extern "C" void kernel_launch(void* const* d_in, const int* in_sizes, int n_in,
                              void* d_out, int out_size, void* d_ws, size_t ws_size,
                              hipStream_t stream) {
  const float* AT = (const float*)d_in[0];
  const float* K_raw = (const float*)d_in[1];
  const float* BT_raw = (const float*)d_in[2];
  const float* W_raw = (const float*)d_in[3];
  const float* b_raw = (const float*)d_in[4];
  float* Y = (float*)d_out;

  const int B = in_sizes[0] / 64;  // 16384
  const int blocks = B / 128;      // 8 waves * 16 rows per block
  comp_net_kernel<<<blocks, 256, 0, stream>>>(AT, K_raw, BT_raw, W_raw, b_raw, Y);
}